// QGraphSAGE_75350906241116
// MI455X (gfx1250) — compile-verified
//
#include <hip/hip_runtime.h>

typedef __attribute__((ext_vector_type(2))) float v2f;
typedef __attribute__((ext_vector_type(8))) float v8f;

#define F_IN  128
#define F_H   256
#define F_OUT 64
#define N1    20000
#define N2    4000

// Workspace layout (float offsets)
#define OFF_AGG1 ((size_t)0)                         // N1*F_IN   = 2,560,000
#define OFF_DEG1 ((size_t)2560000)                   // N1        = 20,000
#define OFF_H    ((size_t)2580000)                   // N1*F_H    = 5,120,000
#define OFF_AGG2 ((size_t)7700000)                   // N2*F_H    = 1,024,000
#define OFF_DEG2 ((size_t)8724000)                   // N2        = 4,000
#define OFF_STAT ((size_t)8728000)                   // 2 floats (sum, sumsq)
#define WS_FLOATS ((size_t)8728004)

// ---------------------------------------------------------------------------
// Zero the accumulation regions of the workspace (h is fully overwritten).
// ---------------------------------------------------------------------------
__global__ void zero_ws_kernel(float* __restrict__ ws, size_t n) {
    size_t i = (size_t)blockIdx.x * blockDim.x + threadIdx.x;
    size_t stride = (size_t)gridDim.x * blockDim.x;
    for (; i < n; i += stride) ws[i] = 0.0f;
}

// ---------------------------------------------------------------------------
// Layer-1 edge scatter: agg1[dst] += x[src]; deg1[dst] += 1.
// One thread per (edge, 4-float chunk): float4 gather + 4 f32 atomics (L2 RMW).
// ---------------------------------------------------------------------------
__global__ void scatter1_kernel(const float* __restrict__ x,
                                const int* __restrict__ src,
                                const int* __restrict__ dst,
                                float* __restrict__ agg,
                                float* __restrict__ deg,
                                int E) {
    int tid = blockIdx.x * blockDim.x + threadIdx.x;
    int e = tid >> 5;                 // F_IN/4 = 32 chunks per edge
    int j = tid & 31;
    if (e >= E) return;
    int s = src[e], d = dst[e];
    float4 v = reinterpret_cast<const float4*>(x + (size_t)s * F_IN)[j];
    float* a = agg + (size_t)d * F_IN + 4 * j;
    atomicAdd(a + 0, v.x);
    atomicAdd(a + 1, v.y);
    atomicAdd(a + 2, v.z);
    atomicAdd(a + 3, v.w);
    if (j == 0) atomicAdd(deg + d, 1.0f);
}

// ---------------------------------------------------------------------------
// Layer-1 GEMM via V_WMMA_F32_16X16X4_F32 (exact fp32).
// One wave32 computes one 16x16 tile of h = x*Wself^T + (agg/deg)*Wneigh^T + b,
// applies ReLU, stores h, and accumulates global sum/sumsq (wave-reduced).
// A frag layout (16x4 f32): lanes 0-15 hold K=0,1; lanes 16-31 hold K=2,3
//   -> contiguous float2 load per lane.  B frag mirrors with N=lane%16.
// ---------------------------------------------------------------------------
__global__ void gemm1_kernel(const float* __restrict__ x,
                             const float* __restrict__ agg,
                             const float* __restrict__ deg,
                             const float* __restrict__ Wself,
                             const float* __restrict__ Wneigh,
                             const float* __restrict__ bias,
                             float* __restrict__ h,
                             float* __restrict__ stats) {
    const int wave = threadIdx.x >> 5;
    const int lane = threadIdx.x & 31;
    const int tile = blockIdx.x * (blockDim.x >> 5) + wave;
    const int NT_N = F_H / 16;                 // 16 N-tiles
    const int mt = tile / NT_N;
    const int nt = tile % NT_N;
    const int r = lane & 15;
    const int half = lane >> 4;
    const int m0 = mt * 16, n0 = nt * 16;

    v8f c = {0.f, 0.f, 0.f, 0.f, 0.f, 0.f, 0.f, 0.f};

    // ---- self contribution: x[m0..m0+15] * Wself^T ----
    {
        const float* Arow = x + (size_t)(m0 + r) * F_IN + 2 * half;
        const float* Brow = Wself + (size_t)(n0 + r) * F_IN + 2 * half;
        for (int k0 = 0; k0 < F_IN; k0 += 4) {
            v2f a, b;
            a[0] = Arow[k0];  a[1] = Arow[k0 + 1];
            b[0] = Brow[k0];  b[1] = Brow[k0 + 1];
            c = __builtin_amdgcn_wmma_f32_16x16x4_f32(false, a, false, b,
                                                      (short)0, c, false, false);
        }
    }
    // ---- neighbor contribution: (agg/max(deg,1)) * Wneigh^T ----
    {
        const float dinv = 1.0f / fmaxf(deg[m0 + r], 1.0f);
        const float* Arow = agg + (size_t)(m0 + r) * F_IN + 2 * half;
        const float* Brow = Wneigh + (size_t)(n0 + r) * F_IN + 2 * half;
        for (int k0 = 0; k0 < F_IN; k0 += 4) {
            v2f a, b;
            a[0] = Arow[k0] * dinv;  a[1] = Arow[k0 + 1] * dinv;
            b[0] = Brow[k0];         b[1] = Brow[k0 + 1];
            c = __builtin_amdgcn_wmma_f32_16x16x4_f32(false, a, false, b,
                                                      (short)0, c, false, false);
        }
    }

    // ---- bias + ReLU + store + global-LN statistics ----
    // C/D layout: VGPR i -> row m0 + 8*half + i, col n0 + r
    const float bn = bias[n0 + r];
    float s = 0.f, sq = 0.f;
    #pragma unroll
    for (int i = 0; i < 8; ++i) {
        const int row = m0 + 8 * half + i;
        float v = fmaxf(c[i] + bn, 0.0f);
        h[(size_t)row * F_H + n0 + r] = v;
        s += v;
        sq += v * v;
    }
    #pragma unroll
    for (int off = 16; off >= 1; off >>= 1) {
        s  += __shfl_xor(s,  off, 32);
        sq += __shfl_xor(sq, off, 32);
    }
    if (lane == 0) {
        atomicAdd(&stats[0], s);
        atomicAdd(&stats[1], sq);
    }
}

// ---------------------------------------------------------------------------
// Layer-2 edge scatter with fused global normalization:
//   agg2[dst] += (h[src] - mu) * rs;  deg2[dst] += 1.
// ---------------------------------------------------------------------------
__global__ void scatter2_kernel(const float* __restrict__ h,
                                const int* __restrict__ src,
                                const int* __restrict__ dst,
                                const float* __restrict__ stats,
                                float* __restrict__ agg,
                                float* __restrict__ deg,
                                int E) {
    int tid = blockIdx.x * blockDim.x + threadIdx.x;
    int e = tid >> 6;                 // F_H/4 = 64 chunks per edge
    int j = tid & 63;
    if (e >= E) return;

    const float cnt = (float)N1 * (float)F_H;
    const float mu = stats[0] / cnt;
    const float var = stats[1] / cnt - mu * mu;
    const float rs = rsqrtf(var + 1e-5f);

    int s = src[e], d = dst[e];
    float4 v = reinterpret_cast<const float4*>(h + (size_t)s * F_H)[j];
    float* a = agg + (size_t)d * F_H + 4 * j;
    atomicAdd(a + 0, (v.x - mu) * rs);
    atomicAdd(a + 1, (v.y - mu) * rs);
    atomicAdd(a + 2, (v.z - mu) * rs);
    atomicAdd(a + 3, (v.w - mu) * rs);
    if (j == 0) atomicAdd(deg + d, 1.0f);
}

// ---------------------------------------------------------------------------
// Layer-2 GEMM via V_WMMA_F32_16X16X4_F32, normalization fused into the
// self-path A loads (agg2 already holds normalized sums).
// out = hn[:4000]*Wself2^T + (agg2/max(deg2,1))*Wneigh2^T + b2
// ---------------------------------------------------------------------------
__global__ void gemm2_kernel(const float* __restrict__ h,
                             const float* __restrict__ agg,
                             const float* __restrict__ deg,
                             const float* __restrict__ stats,
                             const float* __restrict__ Wself,
                             const float* __restrict__ Wneigh,
                             const float* __restrict__ bias,
                             float* __restrict__ out) {
    const int wave = threadIdx.x >> 5;
    const int lane = threadIdx.x & 31;
    const int tile = blockIdx.x * (blockDim.x >> 5) + wave;
    const int NT_N = F_OUT / 16;               // 4 N-tiles
    const int mt = tile / NT_N;
    const int nt = tile % NT_N;
    const int r = lane & 15;
    const int half = lane >> 4;
    const int m0 = mt * 16, n0 = nt * 16;

    const float cnt = (float)N1 * (float)F_H;
    const float mu = stats[0] / cnt;
    const float var = stats[1] / cnt - mu * mu;
    const float rs = rsqrtf(var + 1e-5f);

    v8f c = {0.f, 0.f, 0.f, 0.f, 0.f, 0.f, 0.f, 0.f};

    // ---- self contribution with fused normalization ----
    {
        const float* Arow = h + (size_t)(m0 + r) * F_H + 2 * half;
        const float* Brow = Wself + (size_t)(n0 + r) * F_H + 2 * half;
        for (int k0 = 0; k0 < F_H; k0 += 4) {
            v2f a, b;
            a[0] = (Arow[k0] - mu) * rs;  a[1] = (Arow[k0 + 1] - mu) * rs;
            b[0] = Brow[k0];              b[1] = Brow[k0 + 1];
            c = __builtin_amdgcn_wmma_f32_16x16x4_f32(false, a, false, b,
                                                      (short)0, c, false, false);
        }
    }
    // ---- neighbor contribution (agg2 already normalized) ----
    {
        const float dinv = 1.0f / fmaxf(deg[m0 + r], 1.0f);
        const float* Arow = agg + (size_t)(m0 + r) * F_H + 2 * half;
        const float* Brow = Wneigh + (size_t)(n0 + r) * F_H + 2 * half;
        for (int k0 = 0; k0 < F_H; k0 += 4) {
            v2f a, b;
            a[0] = Arow[k0] * dinv;  a[1] = Arow[k0 + 1] * dinv;
            b[0] = Brow[k0];         b[1] = Brow[k0 + 1];
            c = __builtin_amdgcn_wmma_f32_16x16x4_f32(false, a, false, b,
                                                      (short)0, c, false, false);
        }
    }

    const float bn = bias[n0 + r];
    #pragma unroll
    for (int i = 0; i < 8; ++i) {
        const int row = m0 + 8 * half + i;
        out[(size_t)row * F_OUT + n0 + r] = c[i] + bn;
    }
}

// ---------------------------------------------------------------------------
extern "C" void kernel_launch(void* const* d_in, const int* in_sizes, int n_in,
                              void* d_out, int out_size, void* d_ws, size_t ws_size,
                              hipStream_t stream) {
    const float* x       = (const float*)d_in[0];
    const float* Wself1  = (const float*)d_in[1];
    const float* Wneigh1 = (const float*)d_in[2];
    const float* b1      = (const float*)d_in[3];
    const float* Wself2  = (const float*)d_in[4];
    const float* Wneigh2 = (const float*)d_in[5];
    const float* b2      = (const float*)d_in[6];
    const int*   src1    = (const int*)d_in[7];
    const int*   dst1    = (const int*)d_in[8];
    const int*   src2    = (const int*)d_in[9];
    const int*   dst2    = (const int*)d_in[10];
    const int E1 = in_sizes[7];
    const int E2 = in_sizes[9];

    float* ws    = (float*)d_ws;
    float* agg1  = ws + OFF_AGG1;
    float* deg1  = ws + OFF_DEG1;
    float* h     = ws + OFF_H;
    float* agg2  = ws + OFF_AGG2;
    float* deg2  = ws + OFF_DEG2;
    float* stats = ws + OFF_STAT;
    float* out   = (float*)d_out;

    // 1) zero accumulators + stats
    zero_ws_kernel<<<2048, 256, 0, stream>>>(ws, WS_FLOATS);

    // 2) layer-1 edge scatter
    {
        long long threads = (long long)E1 * 32;
        int blocks = (int)((threads + 255) / 256);
        scatter1_kernel<<<blocks, 256, 0, stream>>>(x, src1, dst1, agg1, deg1, E1);
    }

    // 3) layer-1 WMMA GEMM + ReLU + LN stats (20000 tiles, 8 waves/block)
    {
        int tiles = (N1 / 16) * (F_H / 16);    // 20000
        gemm1_kernel<<<tiles / 8, 256, 0, stream>>>(x, agg1, deg1, Wself1, Wneigh1,
                                                    b1, h, stats);
    }

    // 4) layer-2 edge scatter with fused normalization
    {
        long long threads = (long long)E2 * 64;
        int blocks = (int)((threads + 255) / 256);
        scatter2_kernel<<<blocks, 256, 0, stream>>>(h, src2, dst2, stats,
                                                    agg2, deg2, E2);
    }

    // 5) layer-2 WMMA GEMM -> d_out (1000 tiles, 8 waves/block)
    {
        int tiles = (N2 / 16) * (F_OUT / 16);  // 1000
        gemm2_kernel<<<tiles / 8, 256, 0, stream>>>(h, agg2, deg2, stats,
                                                    Wself2, Wneigh2, b2, out);
    }
}